// GATModelWPathways_89541478187030
// MI455X (gfx1250) — compile-verified
//
#include <hip/hip_runtime.h>
#include <hip/hip_bf16.h>
#include <math.h>

#define N_NODESC 15135
#define N_CMTC   500
#define BS       4
#define HEADS    8
#define HID      8
#define HC       64
#define NEG_SLOPE 0.2f

typedef __attribute__((ext_vector_type(2))) float v2f;
typedef __attribute__((ext_vector_type(8))) float v8f;

#if defined(__has_builtin)
#if __has_builtin(__builtin_amdgcn_wmma_f32_16x16x4_f32)
#define HAVE_WMMA_F32X4 1
#endif
#endif

// ---------------------------------------------------------------------------
// GEMM: H[R,64] = X[R,K] @ W[K,64] using V_WMMA_F32_16X16X4_F32.
// One wave computes a 16-row x 64-col tile (4 accumulators of v8f).
// W is staged once per block into LDS as packed {W[k][c], W[k+1][c]} pairs so
// each B fragment is a single ds_load_b64; A rows are clamped (not predicated)
// so the A fragment is one unconditional global_load_b64 and EXEC stays
// all-ones around the WMMAs.
// ---------------------------------------------------------------------------
__global__ void __launch_bounds__(256) k_gemm(const float* __restrict__ X,
                                              const float* __restrict__ W,
                                              float* __restrict__ H, int R, int K) {
  __shared__ v2f Wl[(HC / 2) * HC];          // up to 64x64 floats packed in k-pairs

  // Cooperative stage of W: Wl[kp*64 + col] = {W[2kp][col], W[2kp+1][col]}
  const int npairs = (K >> 1) * HC;
  for (int idx = threadIdx.x; idx < npairs; idx += 256) {
    const int kp = idx >> 6, col = idx & 63;
    v2f w;
    w.x = W[(size_t)(2 * kp) * HC + col];
    w.y = W[(size_t)(2 * kp + 1) * HC + col];
    Wl[idx] = w;
  }
  __syncthreads();

  const int wave = threadIdx.x >> 5;
  const int lane = threadIdx.x & 31;
  const int tile = blockIdx.x * 8 + wave;
  const int ntiles = (R + 15) >> 4;
  if (tile >= ntiles) return;               // wave-uniform
  const int row0 = tile << 4;
  const int half = lane >> 4;               // 0: K pair {k0,k0+1}, 1: {k0+2,k0+3}
  const int l16  = lane & 15;

#if HAVE_WMMA_F32X4
  // Clamp instead of predicate: garbage rows are dropped at the store.
  const int arow = min(row0 + l16, R - 1);
  const float* __restrict__ arowp = X + (size_t)arow * K;

  v8f acc[4];
#pragma unroll
  for (int ct = 0; ct < 4; ++ct) acc[ct] = v8f{0, 0, 0, 0, 0, 0, 0, 0};

  for (int k0 = 0; k0 < K; k0 += 4) {
    const int ka = k0 + half * 2;           // even
    const v2f a = *(const v2f*)(arowp + ka);
    const v2f* __restrict__ brow = Wl + (ka >> 1) * HC;
#pragma unroll
    for (int ct = 0; ct < 4; ++ct) {
      const v2f b = brow[ct * 16 + l16];
      acc[ct] = __builtin_amdgcn_wmma_f32_16x16x4_f32(
          false, a, false, b, (short)0, acc[ct], false, false);
    }
  }
#pragma unroll
  for (int ct = 0; ct < 4; ++ct) {
#pragma unroll
    for (int r = 0; r < 8; ++r) {
      const int mrow = row0 + r + half * 8;   // C/D layout: VGPR r -> M=r / M=r+8
      if (mrow < R) H[(size_t)mrow * HC + ct * 16 + l16] = acc[ct][r];
    }
  }
#else
  // scalar fallback (should not trigger on gfx1250)
  for (int idx = lane; idx < 16 * HC; idx += 32) {
    const int r = idx >> 6, c = idx & 63;
    const int row = row0 + r;
    if (row >= R) continue;
    float s = 0.f;
    for (int k = 0; k < K; ++k) s += X[(size_t)row * K + k] * W[(size_t)k * HC + c];
    H[(size_t)row * HC + c] = s;
  }
#endif
}

// al_src/al_dst: per (row, head) dot of h[row, head,:] with a_src/a_dst[head,:]
__global__ void k_node_alpha(const float* __restrict__ H,
                             const float* __restrict__ a_src,
                             const float* __restrict__ a_dst,
                             float* __restrict__ alS, float* __restrict__ alD, int R) {
  const int t = blockIdx.x * blockDim.x + threadIdx.x;
  if (t >= R * HEADS) return;
  const int row = t >> 3, hd = t & 7;
  const float* hp = H + (size_t)row * HC + hd * HID;
  const float* as = a_src + hd * HID;
  const float* ad = a_dst + hd * HID;
  float s = 0.f, d = 0.f;
#pragma unroll
  for (int c = 0; c < HID; ++c) { s += hp[c] * as[c]; d += hp[c] * ad[c]; }
  alS[t] = s; alD[t] = d;
}

__global__ void k_init_md(float* __restrict__ m, float* __restrict__ den, int n) {
  const int t = blockIdx.x * blockDim.x + threadIdx.x;
  if (t < n) { m[t] = -INFINITY; den[t] = 0.f; }
}

__global__ void k_zero(float* __restrict__ p, int n) {
  const int t = blockIdx.x * blockDim.x + threadIdx.x;
  if (t < n) p[t] = 0.f;
}

__device__ __forceinline__ void edge_sd(const int* __restrict__ ei, int E, int e,
                                        int& s, int& d) {
  if (e < E) { s = ei[e]; d = ei[E + e]; } else { s = e - E; d = e - E; }
}

__device__ __forceinline__ float edge_logit(const float* __restrict__ alS,
                                            const float* __restrict__ alD,
                                            int s, int d, int b, int hd) {
  const float v = alS[((size_t)b * N_NODESC + s) * HEADS + hd] +
                  alD[((size_t)b * N_NODESC + d) * HEADS + hd];
  return v > 0.f ? v : NEG_SLOPE * v;
}

__device__ __forceinline__ void atomicMaxF(float* addr, float v) {
  if (v >= 0.f) atomicMax((int*)addr, __float_as_int(v));
  else          atomicMin((unsigned int*)addr, __float_as_uint(v));
}

// thread per (edge, b, head)
__global__ void k_edge_max(const int* __restrict__ ei, int E, int Etot,
                           const float* __restrict__ alS, const float* __restrict__ alD,
                           float* __restrict__ m) {
  const int t = blockIdx.x * blockDim.x + threadIdx.x;
  if (t >= Etot * (BS * HEADS)) return;
  const int e = t >> 5, rem = t & 31, b = rem >> 3, hd = rem & 7;
  int s, d; edge_sd(ei, E, e, s, d);
  const float v = edge_logit(alS, alD, s, d, b, hd);
  atomicMaxF(&m[(size_t)d * (BS * HEADS) + rem], v);
}

__global__ void k_edge_sum(const int* __restrict__ ei, int E, int Etot,
                           const float* __restrict__ alS, const float* __restrict__ alD,
                           const float* __restrict__ m, float* __restrict__ den) {
  const int t = blockIdx.x * blockDim.x + threadIdx.x;
  if (t >= Etot * (BS * HEADS)) return;
  const int e = t >> 5, rem = t & 31, b = rem >> 3, hd = rem & 7;
  int s, d; edge_sd(ei, E, e, s, d);
  const float v = edge_logit(alS, alD, s, d, b, hd);
  const float ex = __expf(v - m[(size_t)d * (BS * HEADS) + rem]);
  atomicAdd(&den[(size_t)d * (BS * HEADS) + rem], ex);
}

// alpha recomputed; 8 channel atomic adds into acc[b][dst][hd*8+c]
__global__ void k_edge_msg(const int* __restrict__ ei, int E, int Etot,
                           const float* __restrict__ alS, const float* __restrict__ alD,
                           const float* __restrict__ m, const float* __restrict__ den,
                           const float* __restrict__ H, float* __restrict__ acc) {
  const int t = blockIdx.x * blockDim.x + threadIdx.x;
  if (t >= Etot * (BS * HEADS)) return;
  const int e = t >> 5, rem = t & 31, b = rem >> 3, hd = rem & 7;
  int s, d; edge_sd(ei, E, e, s, d);
  const float v = edge_logit(alS, alD, s, d, b, hd);
  const size_t mi = (size_t)d * (BS * HEADS) + rem;
  const float alpha = __expf(v - m[mi]) / den[mi];
  const float* hs = H + ((size_t)b * N_NODESC + s) * HC + hd * HID;
  float* ap = acc + ((size_t)b * N_NODESC + d) * HC + hd * HID;
#pragma unroll
  for (int c = 0; c < HID; ++c) atomicAdd(&ap[c], alpha * hs[c]);
}

__global__ void k_finalize(const float* __restrict__ acc, const float* __restrict__ bvec,
                           float* __restrict__ Xout, int n) {
  const int t = blockIdx.x * blockDim.x + threadIdx.x;
  if (t >= n) return;
  const float v = acc[t] + bvec[t & (HC - 1)];
  Xout[t] = v > 0.f ? v : 0.f;
}

// pathway pooling: thread per (p, b); xc interleaves layers: f = k*3 + l
__global__ void k_pool_gather(const int* __restrict__ prow, const int* __restrict__ pcol,
                              const float* __restrict__ X1, const float* __restrict__ X2,
                              const float* __restrict__ X3, float* __restrict__ sums, int P) {
  const int t = blockIdx.x * blockDim.x + threadIdx.x;
  if (t >= P * BS) return;
  const int p = t >> 2, b = t & 3;
  const int row = prow[p], cmt = pcol[p];
  const size_t nb = ((size_t)b * N_NODESC + row) * HC;
  float* out = sums + ((size_t)b * N_CMTC + cmt) * (3 * HC);
#pragma unroll 4
  for (int k = 0; k < HC; ++k) {
    atomicAdd(&out[k * 3 + 0], X1[nb + k]);
    atomicAdd(&out[k * 3 + 1], X2[nb + k]);
    atomicAdd(&out[k * 3 + 2], X3[nb + k]);
  }
}

__global__ void k_pool_cnt(const int* __restrict__ pcol, float* __restrict__ cnt, int P) {
  const int t = blockIdx.x * blockDim.x + threadIdx.x;
  if (t < P) atomicAdd(&cnt[pcol[t]], 1.f);
}

// y[b,cmt] = (pooled/cnt) . fc_w + fc_b
__global__ void k_fc(const float* __restrict__ sums, const float* __restrict__ cnt,
                     const float* __restrict__ fcw, const float* __restrict__ fcb,
                     float* __restrict__ y) {
  const int t = blockIdx.x * blockDim.x + threadIdx.x;
  if (t >= BS * N_CMTC) return;
  const int b = t / N_CMTC, cm = t % N_CMTC;
  const float inv = 1.f / fmaxf(cnt[cm], 1.f);
  const float* sp = sums + ((size_t)b * N_CMTC + cm) * (3 * HC);
  float a = fcb[0];
  for (int f = 0; f < 3 * HC; ++f) a += sp[f] * inv * fcw[f];
  y[t] = a;
}

__global__ void k_lin1(const float* __restrict__ y, const float* __restrict__ w,
                       const float* __restrict__ b, float* __restrict__ h1) {
  const int t = blockIdx.x * blockDim.x + threadIdx.x;
  if (t >= BS * 128) return;
  const int bb = t >> 7, j = t & 127;
  float a = b[j];
  for (int c = 0; c < N_CMTC; ++c) a += y[bb * N_CMTC + c] * w[c * 128 + j];
  h1[t] = a > 0.f ? a : 0.f;
}

__global__ void k_head(const float* __restrict__ h1, const float* __restrict__ w,
                       const float* __restrict__ b, float* __restrict__ out) {
  __shared__ float z[BS * 2];
  const int t = threadIdx.x;
  if (t < BS * 2) {
    const int bb = t >> 1, c = t & 1;
    float a = b[c];
    for (int j = 0; j < 128; ++j) a += h1[bb * 128 + j] * w[j * 2 + c];
    z[t] = a;
  }
  __syncthreads();
  if (t < BS * 2) {
    const int bb = t >> 1;
    const float mx = fmaxf(z[bb * 2], z[bb * 2 + 1]);
    out[t] = z[t] - mx - logf(expf(z[bb * 2] - mx) + expf(z[bb * 2 + 1] - mx));
  }
}

// ---------------------------------------------------------------------------
static inline int cdiv(int a, int b) { return (a + b - 1) / b; }

static void run_gat_layer(hipStream_t stream, const float* Xin, int K,
                          const float* W, const float* as, const float* ad,
                          const float* bvec, const int* ei, int E, int Etot,
                          float* h, float* acc, float* alS, float* alD,
                          float* m, float* den, float* Xout) {
  const int R = BS * N_NODESC;
  const int ntiles = (R + 15) >> 4;
  k_gemm<<<cdiv(ntiles, 8), 256, 0, stream>>>(Xin, W, h, R, K);
  k_node_alpha<<<cdiv(R * HEADS, 256), 256, 0, stream>>>(h, as, ad, alS, alD, R);
  k_init_md<<<cdiv(N_NODESC * BS * HEADS, 256), 256, 0, stream>>>(m, den, N_NODESC * BS * HEADS);
  k_zero<<<cdiv(R * HC, 256), 256, 0, stream>>>(acc, R * HC);
  const int ET = Etot * BS * HEADS;
  k_edge_max<<<cdiv(ET, 256), 256, 0, stream>>>(ei, E, Etot, alS, alD, m);
  k_edge_sum<<<cdiv(ET, 256), 256, 0, stream>>>(ei, E, Etot, alS, alD, m, den);
  k_edge_msg<<<cdiv(ET, 256), 256, 0, stream>>>(ei, E, Etot, alS, alD, m, den, h, acc);
  k_finalize<<<cdiv(R * HC, 256), 256, 0, stream>>>(acc, bvec, Xout, R * HC);
}

extern "C" void kernel_launch(void* const* d_in, const int* in_sizes, int n_in,
                              void* d_out, int out_size, void* d_ws, size_t ws_size,
                              hipStream_t stream) {
  const float* x     = (const float*)d_in[0];
  const int*   ei    = (const int*)d_in[1];
  const int*   prow  = (const int*)d_in[2];
  const int*   pcol  = (const int*)d_in[3];
  const float* W1    = (const float*)d_in[4];
  const float* as1   = (const float*)d_in[5];
  const float* ad1   = (const float*)d_in[6];
  const float* b1    = (const float*)d_in[7];
  const float* W2    = (const float*)d_in[8];
  const float* as2   = (const float*)d_in[9];
  const float* ad2   = (const float*)d_in[10];
  const float* b2    = (const float*)d_in[11];
  const float* W3    = (const float*)d_in[12];
  const float* as3   = (const float*)d_in[13];
  const float* ad3   = (const float*)d_in[14];
  const float* b3    = (const float*)d_in[15];
  const float* fcw   = (const float*)d_in[16];
  const float* fcb   = (const float*)d_in[17];
  const float* l1w   = (const float*)d_in[18];
  const float* l1b   = (const float*)d_in[19];
  const float* l2w   = (const float*)d_in[20];
  const float* l2b   = (const float*)d_in[21];

  const int E = in_sizes[1] / 2;
  const int P = in_sizes[2];
  const int Etot = E + N_NODESC;
  const int R = BS * N_NODESC;

  float* ws = (float*)d_ws;
  size_t o = 0;
  float* h    = ws + o; o += (size_t)R * HC;
  float* acc  = ws + o; o += (size_t)R * HC;
  float* x1   = ws + o; o += (size_t)R * HC;
  float* x2   = ws + o; o += (size_t)R * HC;
  float* x3   = ws + o; o += (size_t)R * HC;
  float* alS  = ws + o; o += (size_t)R * HEADS;
  float* alD  = ws + o; o += (size_t)R * HEADS;
  float* m    = ws + o; o += (size_t)N_NODESC * BS * HEADS;
  float* den  = ws + o; o += (size_t)N_NODESC * BS * HEADS;
  float* sums = ws + o; o += (size_t)BS * N_CMTC * 3 * HC;
  float* cnt  = ws + o; o += 512;
  float* y    = ws + o; o += (size_t)BS * N_CMTC;
  float* h1   = ws + o; o += (size_t)BS * 128;

  run_gat_layer(stream, x,  8,  W1, as1, ad1, b1, ei, E, Etot, h, acc, alS, alD, m, den, x1);
  run_gat_layer(stream, x1, HC, W2, as2, ad2, b2, ei, E, Etot, h, acc, alS, alD, m, den, x2);
  run_gat_layer(stream, x2, HC, W3, as3, ad3, b3, ei, E, Etot, h, acc, alS, alD, m, den, x3);

  k_zero<<<cdiv(BS * N_CMTC * 3 * HC, 256), 256, 0, stream>>>(sums, BS * N_CMTC * 3 * HC);
  k_zero<<<cdiv(512, 256), 256, 0, stream>>>(cnt, 512);
  k_pool_gather<<<cdiv(P * BS, 256), 256, 0, stream>>>(prow, pcol, x1, x2, x3, sums, P);
  k_pool_cnt<<<cdiv(P, 256), 256, 0, stream>>>(pcol, cnt, P);
  k_fc<<<cdiv(BS * N_CMTC, 256), 256, 0, stream>>>(sums, cnt, fcw, fcb, y);
  k_lin1<<<cdiv(BS * 128, 256), 256, 0, stream>>>(y, l1w, l1b, h1);
  k_head<<<1, 32, 0, stream>>>(h1, l2w, l2b, (float*)d_out);
}